// NET_2216203125270
// MI455X (gfx1250) — compile-verified
//
#include <hip/hip_runtime.h>
#include <math.h>

#define D 128

typedef __attribute__((ext_vector_type(2))) float v2f;
typedef __attribute__((ext_vector_type(8))) float v8f;

// ---------------------------------------------------------------------------
// Zero a float buffer (needed every call: workspace is poisoned / not reset).
// ---------------------------------------------------------------------------
__global__ void zero_kernel(float* __restrict__ p, int n) {
    int i = blockIdx.x * blockDim.x + threadIdx.x;
    if (i < n) p[i] = 0.0f;
}

// ---------------------------------------------------------------------------
// Degree counting: out_deg over src, in_deg over dst (float atomics).
// ---------------------------------------------------------------------------
__global__ void deg_kernel(const int* __restrict__ src, const int* __restrict__ dst,
                           float* __restrict__ deg_src, float* __restrict__ deg_dst, int E) {
    int i = blockIdx.x * blockDim.x + threadIdx.x;
    if (i < E) {
        atomicAdd(&deg_src[src[i]], 1.0f);
        atomicAdd(&deg_dst[dst[i]], 1.0f);
    }
}

// deg -> rsqrt(max(deg,1)) in place (covers both deg buffers, contiguous).
__global__ void norm_kernel(float* __restrict__ deg, int n) {
    int i = blockIdx.x * blockDim.x + threadIdx.x;
    if (i < n) deg[i] = rsqrtf(fmaxf(deg[i], 1.0f));
}

// ---------------------------------------------------------------------------
// hs = h * norm_src (row-broadcast); agg = 0.  Vectorized float4.
// n4 = N*D/4; D/4 = 32 float4 per row.
// ---------------------------------------------------------------------------
__global__ void scale_zero_kernel(const float4* __restrict__ h4,
                                  const float* __restrict__ norm_src,
                                  float4* __restrict__ hs4,
                                  float4* __restrict__ agg4, int n4) {
    int i = blockIdx.x * blockDim.x + threadIdx.x;
    if (i < n4) {
        float s = norm_src[i >> 5];            // row = (i*4)/128
        float4 v = h4[i];
        v.x *= s; v.y *= s; v.z *= s; v.w *= s;
        hs4[i] = v;
        agg4[i] = make_float4(0.0f, 0.0f, 0.0f, 0.0f);
    }
}

// ---------------------------------------------------------------------------
// Scatter-add: one wave32 per edge. Lane l owns 16B chunk l of the 512B row.
// agg[dst] += hs[src] * w.  Edge scalars are wave-uniform -> scalarized.
// ---------------------------------------------------------------------------
__global__ __launch_bounds__(256) void scatter_kernel(const float* __restrict__ hs,
                                                      const int* __restrict__ src,
                                                      const int* __restrict__ dst,
                                                      const float* __restrict__ ew,
                                                      float* __restrict__ agg, int E) {
    int wave = (blockIdx.x * blockDim.x + threadIdx.x) >> 5;
    int lane = threadIdx.x & 31;
    if (wave >= E) return;
    int s = src[wave];
    int d = dst[wave];
    float w = ew[wave];
    const float4* row = (const float4*)(hs + (size_t)s * D);
    float4 v = row[lane];
    float* o = agg + (size_t)d * D + lane * 4;
    atomicAdd(o + 0, v.x * w);
    atomicAdd(o + 1, v.y * w);
    atomicAdd(o + 2, v.z * w);
    atomicAdd(o + 3, v.w * w);
}

// ---------------------------------------------------------------------------
// out = relu(relu((agg @ W) * norm_dst[row] + b[col]))
// One block = 16-row strip; 8 waves = 8 column tiles of 16; K in steps of 4
// via V_WMMA_F32_16X16X4_F32 (native fp32 WMMA).
// A-tile (16x128, 8KB) staged in LDS; W served from L2/WGP$.
// ISA layouts:
//   A 16x4: lane<16 -> M=lane,    VGPR{0,1}=K{0,1};  lane>=16 -> M=lane-16, K{2,3}
//   B 4x16: lane<16 -> N=lane,    VGPR{0,1}=K{0,1};  lane>=16 -> N=lane-16, K{2,3}
//   C/D 16x16: VGPR v -> row v (lanes<16) / row v+8 (lanes>=16), col = lane&15
// ---------------------------------------------------------------------------
__global__ __launch_bounds__(256) void gemm_norm_relu_kernel(const float* __restrict__ agg,
                                                             const float* __restrict__ W,
                                                             const float* __restrict__ bias,
                                                             const float* __restrict__ norm_dst,
                                                             float* __restrict__ out) {
    __shared__ float sA[16 * D];               // 8 KB
    const int rowBase = blockIdx.x * 16;

    // Cooperative load of the 16x128 A strip: 2048 floats = 512 float4.
    {
        const float4* g = (const float4*)(agg + (size_t)rowBase * D);
        float4* s4 = (float4*)sA;
        s4[threadIdx.x]       = g[threadIdx.x];
        s4[threadIdx.x + 256] = g[threadIdx.x + 256];
    }
    __syncthreads();

    const int waveId  = threadIdx.x >> 5;      // 0..7 -> column tile
    const int lane    = threadIdx.x & 31;
    const int colBase = waveId * 16;
    const int col     = colBase + (lane & 15);
    const int khalf   = (lane < 16) ? 0 : 2;   // which K-pair this half-wave owns
    const float* aRow = sA + (lane & 15) * D;

    v8f acc = {};
    #pragma unroll 8
    for (int k = 0; k < D; k += 4) {
        v2f a, b;
        a.x = aRow[k + khalf];
        a.y = aRow[k + khalf + 1];
        b.x = W[(size_t)(k + khalf) * D + col];
        b.y = W[(size_t)(k + khalf + 1) * D + col];
        acc = __builtin_amdgcn_wmma_f32_16x16x4_f32(
                  /*neg_a=*/false, a, /*neg_b=*/false, b,
                  /*c_mod=*/(short)0, acc, /*reuse_a=*/false, /*reuse_b=*/false);
    }

    const float bcol = bias[col];
    const int rowOff = (lane < 16) ? 0 : 8;
    #pragma unroll
    for (int v = 0; v < 8; ++v) {
        int row = rowBase + v + rowOff;
        float r = acc[v] * norm_dst[row] + bcol;
        out[(size_t)row * D + col] = fmaxf(r, 0.0f);   // relu (outer relu idempotent)
    }
}

// ---------------------------------------------------------------------------
// Orchestration. Workspace layout (floats):
//   [0,N)        norm_src   (degree -> rsqrt norm, in place)
//   [N,2N)       norm_dst
//   [2N, +N*D)   hs   (scaled features)
//   [.., +N*D)   agg  (scatter accumulator)
//   [.., +N*D)   h1   (layer-1 output)
// Total = 2N + 3*N*D floats ~= 15.5 MB  (fits comfortably in L2's 192 MB).
// ---------------------------------------------------------------------------
extern "C" void kernel_launch(void* const* d_in, const int* in_sizes, int n_in,
                              void* d_out, int out_size, void* d_ws, size_t ws_size,
                              hipStream_t stream) {
    const float* x   = (const float*)d_in[0];
    const float* ew  = (const float*)d_in[1];
    const float* W0  = (const float*)d_in[2];
    const float* b0  = (const float*)d_in[3];
    const float* W1  = (const float*)d_in[4];
    const float* b1  = (const float*)d_in[5];
    const int*   src = (const int*)d_in[6];
    const int*   dst = (const int*)d_in[7];
    float*       out = (float*)d_out;

    const int N = in_sizes[0] / D;     // 10000
    const int E = in_sizes[1];         // 640000
    const int ND = N * D;
    const int n4 = ND / 4;

    float* ws    = (float*)d_ws;
    float* normS = ws;
    float* normD = ws + N;
    float* hs    = ws + 2 * N;
    float* agg   = hs + (size_t)ND;
    float* h1    = agg + (size_t)ND;

    const int T = 256;

    // Degrees -> norms (shared by both layers).
    zero_kernel<<<(2 * N + T - 1) / T, T, 0, stream>>>(normS, 2 * N);
    deg_kernel<<<(E + T - 1) / T, T, 0, stream>>>(src, dst, normS, normD, E);
    norm_kernel<<<(2 * N + T - 1) / T, T, 0, stream>>>(normS, 2 * N);

    // Layer 0: x -> h1
    scale_zero_kernel<<<(n4 + T - 1) / T, T, 0, stream>>>(
        (const float4*)x, normS, (float4*)hs, (float4*)agg, n4);
    scatter_kernel<<<(E + 7) / 8, T, 0, stream>>>(hs, src, dst, ew, agg, E);
    gemm_norm_relu_kernel<<<N / 16, T, 0, stream>>>(agg, W0, b0, normD, h1);

    // Layer 1: h1 -> out
    scale_zero_kernel<<<(n4 + T - 1) / T, T, 0, stream>>>(
        (const float4*)h1, normS, (float4*)hs, (float4*)agg, n4);
    scatter_kernel<<<(E + 7) / 8, T, 0, stream>>>(hs, src, dst, ew, agg, E);
    gemm_norm_relu_kernel<<<N / 16, T, 0, stream>>>(agg, W1, b1, normD, out);
}